// normal_42090679501125
// MI455X (gfx1250) — compile-verified
//
#include <hip/hip_runtime.h>
#include <math.h>

typedef __attribute__((ext_vector_type(2))) float v2f;
typedef __attribute__((ext_vector_type(8))) float v8f;

#define K_NN 30
#define TILE 16
#define WAVES_PER_BLOCK 8
#define THREADS (WAVES_PER_BLOCK * 32)
#define ROW_PITCH 20  // 16 + 4 pad: 16B-aligned rows, stride not 0 mod 64 banks

// Each wave owns 32 query points. Per 16-candidate tile:
//   A rows (queries): (xq, yq, zq, 1)
//   B cols (cands)  : (-2xc, -2yc, -2zc, |c|^2)
//   => WMMA D[m][n] = |c|^2 - 2 q.c ; lane adds its own |q|^2 for full d2.
// Two v_wmma_f32_16x16x4_f32 per tile (query rows 0-15, 16-31), D bounced
// through LDS so each lane scans its own query row (4x ds_load_b128).
// Candidate coords are software-pipelined (prefetch tile i+1 during tile i).
__global__ __launch_bounds__(THREADS)
void normals_topk_wmma_kernel(const float* __restrict__ xyz,
                              const float* __restrict__ radius_ptr,
                              float* __restrict__ out,
                              int n)
{
    alignas(16) __shared__ float smem[WAVES_PER_BLOCK * 32 * ROW_PITCH];

    const int tid  = threadIdx.x;
    const int wave = tid >> 5;
    const int lane = tid & 31;        // wave32
    const int half = lane >> 4;
    const int l    = lane & 15;

    const int qbase = blockIdx.x * THREADS + wave * 32;
    float* wsm = &smem[wave * 32 * ROW_PITCH];

    // Own query point (top-k ownership: row `lane` of the 32x16 tile).
    const int q = qbase + lane;
    const float qx = xyz[q * 3 + 0];
    const float qy = xyz[q * 3 + 1];
    const float qz = xyz[q * 3 + 2];
    const float sqq = qx * qx + qy * qy + qz * qz;

    // A operands. 16x4 f32 A layout: lanes 0-15 hold {K0,K1} in v[0],v[1];
    // lanes 16-31 hold {K2,K3}. Row m = l. A row = (x, y, z, 1).
    v2f a0, a1;
    {
        const int qa = qbase + l;
        const int qb = qbase + 16 + l;
        const float ax0 = xyz[qa * 3 + 0], ay0 = xyz[qa * 3 + 1], az0 = xyz[qa * 3 + 2];
        const float ax1 = xyz[qb * 3 + 0], ay1 = xyz[qb * 3 + 1], az1 = xyz[qb * 3 + 2];
        a0.x = half ? az0  : ax0;
        a0.y = half ? 1.0f : ay0;
        a1.x = half ? az1  : ax1;
        a1.y = half ? 1.0f : ay1;
    }

    // Register-resident top-K (max-replace scheme).
    float kd[K_NN];
    int   ki[K_NN];
#pragma unroll
    for (int s = 0; s < K_NN; ++s) { kd[s] = 3.0e38f; ki[s] = 0; }
    float wv = 3.0e38f;  // current worst (max) of the K kept distances

    auto scan_one = [&](float dotv, int jidx) {
        const float d2v = sqq + dotv;
        if (d2v < wv) {
            bool done = false;
#pragma unroll
            for (int s = 0; s < K_NN; ++s) {
                const bool take = (!done) && (kd[s] == wv);
                if (take) { kd[s] = d2v; ki[s] = jidx; }
                done = done || take;
            }
            float nm = kd[0];
#pragma unroll
            for (int s = 1; s < K_NN; ++s) nm = fmaxf(nm, kd[s]);
            wv = nm;
        }
    };

    // Prologue: load candidate tile 0 coords for this lane.
    float cx = xyz[l * 3 + 0];
    float cy = xyz[l * 3 + 1];
    float cz = xyz[l * 3 + 2];

    for (int j0 = 0; j0 < n; j0 += TILE) {
        // Prefetch next tile's coords (wraps to 0 on last iter; harmless).
        int jn = j0 + TILE;
        jn = (jn < n) ? jn : 0;
        const int cnex = jn + l;
        const float px = xyz[cnex * 3 + 0];
        const float py = xyz[cnex * 3 + 1];
        const float pz = xyz[cnex * 3 + 2];

        // B operand from current tile coords: col = cand j0+l:
        // (-2x, -2y, -2z, |c|^2), same {vgpr, lane-half} -> K map as A.
        const float sqc = cx * cx + cy * cy + cz * cz;
        v2f b;
        b.x = half ? (-2.0f * cz) : (-2.0f * cx);
        b.y = half ? sqc          : (-2.0f * cy);

        v8f cz8 = {};
        v8f d0 = __builtin_amdgcn_wmma_f32_16x16x4_f32(
            false, a0, false, b, (short)0, cz8, false, false);
        v8f d1 = __builtin_amdgcn_wmma_f32_16x16x4_f32(
            false, a1, false, b, (short)0, cz8, false, false);

        // Scatter D tiles to LDS. D layout: vgpr r = row r (lanes 0-15) or
        // row r+8 (lanes 16-31), col = l. d1 covers query rows 16..31.
#pragma unroll
        for (int r = 0; r < 8; ++r) {
            wsm[(r + half * 8) * ROW_PITCH + l]      = d0[r];
            wsm[(16 + r + half * 8) * ROW_PITCH + l] = d1[r];
        }
        // Same-wave DS ops are in-order; stop the compiler from hoisting
        // the row reads above the tile stores.
        asm volatile("" ::: "memory");

        // Read own query row as 4x b128 (row base is 16B aligned: pitch 20).
        const float4* rowp = (const float4*)&wsm[lane * ROW_PITCH];
        const float4 r0 = rowp[0];
        const float4 r1 = rowp[1];
        const float4 r2 = rowp[2];
        const float4 r3 = rowp[3];

        scan_one(r0.x, j0 + 0);  scan_one(r0.y, j0 + 1);
        scan_one(r0.z, j0 + 2);  scan_one(r0.w, j0 + 3);
        scan_one(r1.x, j0 + 4);  scan_one(r1.y, j0 + 5);
        scan_one(r1.z, j0 + 6);  scan_one(r1.w, j0 + 7);
        scan_one(r2.x, j0 + 8);  scan_one(r2.y, j0 + 9);
        scan_one(r2.z, j0 + 10); scan_one(r2.w, j0 + 11);
        scan_one(r3.x, j0 + 12); scan_one(r3.y, j0 + 13);
        scan_one(r3.z, j0 + 14); scan_one(r3.w, j0 + 15);

        // Rotate the pipelined candidate coords.
        cx = px; cy = py; cz = pz;
    }

    // ---- Hybrid radius mask, mean ----
    const float rad = radius_ptr[0];
    const float r2v = rad * rad;

    int cnt = 0;
    float mx = 0.f, my = 0.f, mz = 0.f;
#pragma unroll
    for (int s = 0; s < K_NN; ++s) {
        const float w = (kd[s] <= r2v) ? 1.0f : 0.0f;
        const int j = ki[s];
        mx += w * xyz[j * 3 + 0];
        my += w * xyz[j * 3 + 1];
        mz += w * xyz[j * 3 + 2];
        cnt += (int)w;
    }
    const float inv = 1.0f / (float)max(cnt, 1);
    mx *= inv; my *= inv; mz *= inv;

    // ---- Covariance (second gather pass; keeps VGPR pressure sane) ----
    float c00 = 0.f, c01 = 0.f, c02 = 0.f, c11 = 0.f, c12 = 0.f, c22 = 0.f;
#pragma unroll
    for (int s = 0; s < K_NN; ++s) {
        const float w = (kd[s] <= r2v) ? 1.0f : 0.0f;
        const int j = ki[s];
        const float dx = w * (xyz[j * 3 + 0] - mx);
        const float dy = w * (xyz[j * 3 + 1] - my);
        const float dz = w * (xyz[j * 3 + 2] - mz);
        c00 += dx * dx; c01 += dx * dy; c02 += dx * dz;
        c11 += dy * dy; c12 += dy * dz; c22 += dz * dz;
    }
    c00 *= inv; c01 *= inv; c02 *= inv;
    c11 *= inv; c12 *= inv; c22 *= inv;

    // ---- Smallest eigenvalue (trigonometric closed form) ----
    const float p1 = c01 * c01 + c02 * c02 + c12 * c12;
    const float qm = (c00 + c11 + c22) * (1.0f / 3.0f);
    const float e0 = c00 - qm, e1 = c11 - qm, e2 = c22 - qm;
    const float p2 = e0 * e0 + e1 * e1 + e2 * e2 + 2.0f * p1;
    float lmin;
    if (p2 > 1e-30f) {
        const float p = sqrtf(p2 * (1.0f / 6.0f));
        const float ip = 1.0f / p;
        const float b00 = e0 * ip, b11 = e1 * ip, b22 = e2 * ip;
        const float b01 = c01 * ip, b02 = c02 * ip, b12 = c12 * ip;
        float detB = b00 * (b11 * b22 - b12 * b12)
                   - b01 * (b01 * b22 - b12 * b02)
                   + b02 * (b01 * b12 - b11 * b02);
        float rr = 0.5f * detB;
        rr = fminf(1.0f, fmaxf(-1.0f, rr));
        const float phi = acosf(rr) * (1.0f / 3.0f);
        lmin = qm + 2.0f * p * cosf(phi + 2.0943951023931953f);  // +2pi/3
    } else {
        lmin = qm;  // isotropic: eigenvector falls back to (1,0,0) below
    }

    // ---- Eigenvector: largest cross product of rows of (A - lmin*I) ----
    const float r0x = c00 - lmin, r0y = c01,        r0z = c02;
    const float r1x = c01,        r1y = c11 - lmin, r1z = c12;
    const float r2x = c02,        r2y = c12,        r2z = c22 - lmin;

    float ax = r0y * r1z - r0z * r1y, ay = r0z * r1x - r0x * r1z, az = r0x * r1y - r0y * r1x;
    float bx = r0y * r2z - r0z * r2y, by = r0z * r2x - r0x * r2z, bz = r0x * r2y - r0y * r2x;
    float ex = r1y * r2z - r1z * r2y, ey = r1z * r2x - r1x * r2z, ez = r1x * r2y - r1y * r2x;
    const float la = ax * ax + ay * ay + az * az;
    const float lb = bx * bx + by * by + bz * bz;
    const float lc = ex * ex + ey * ey + ez * ez;

    float nx = ax, ny = ay, nz = az, ln = la;
    if (lb > ln) { nx = bx; ny = by; nz = bz; ln = lb; }
    if (lc > ln) { nx = ex; ny = ey; nz = ez; ln = lc; }

    if (ln < 1e-30f) { nx = 1.0f; ny = 0.0f; nz = 0.0f; }
    else {
        const float is = rsqrtf(nx * nx + ny * ny + nz * nz);
        nx *= is; ny *= is; nz *= is;
    }
    if (nz < 0.0f) { nx = -nx; ny = -ny; nz = -nz; }
    if (cnt < 3) { nx = 0.0f; ny = 0.0f; nz = 0.0f; }

    out[q * 3 + 0] = nx;
    out[q * 3 + 1] = ny;
    out[q * 3 + 2] = nz;
}

extern "C" void kernel_launch(void* const* d_in, const int* in_sizes, int n_in,
                              void* d_out, int out_size, void* d_ws, size_t ws_size,
                              hipStream_t stream) {
    (void)n_in; (void)out_size; (void)d_ws; (void)ws_size;
    const float* xyz = (const float*)d_in[0];        // [N,3] f32
    const float* rad = (const float*)d_in[1];        // scalar f32
    float* out = (float*)d_out;                      // [N,3] f32
    const int n = in_sizes[0] / 3;                   // 16384
    const int grid = n / THREADS;                    // 64 blocks of 256 (8 waves)
    normals_topk_wmma_kernel<<<grid, THREADS, 0, stream>>>(xyz, rad, out, n);
}